// GenView_32719060861410
// MI455X (gfx1250) — compile-verified
//
#include <hip/hip_runtime.h>
#include <hip/hip_bf16.h>

typedef __attribute__((ext_vector_type(2))) float v2f;
typedef __attribute__((ext_vector_type(8))) float v8f;

#define HID 64
#define FDIM 256

// ---------- helpers: order-preserving float <-> int key for atomicMax ----------
__device__ __forceinline__ int f2ord(float f) {
    int i = __float_as_int(f);
    return (i >= 0) ? i : (i ^ 0x7FFFFFFF);
}
__device__ __forceinline__ float ord2f(int i) {
    return __int_as_float((i >= 0) ? i : (i ^ 0x7FFFFFFF));
}

// ---------- kernel 0: init workspace (agg=0, z=0, mkey=-inf-key) ----------
__global__ void gv_init_kernel(float* __restrict__ agg, float* __restrict__ z,
                               int* __restrict__ mkey, int N) {
    int idx = blockIdx.x * blockDim.x + threadIdx.x;
    int total = N * HID;
    if (idx < total) agg[idx] = 0.0f;
    if (idx < N) { z[idx] = 0.0f; mkey[idx] = 0x80000000; }  // INT_MIN, below key(-inf)
}

// ---------- kernel 1: h = feat @ W + b  via V_WMMA_F32_16X16X4_F32 ----------
// W (256x64 fp32, 64KB) is staged once per block into LDS in a K-quad-packed,
// bank-conflict-free layout:  Wq[(K>>2)*256 + n*4 + (K&3)] = W[K][n].
//   -> lane reads its (W[K][n], W[K+1][n]) pair as ONE ds_load_b64:
//      lanes 0-15 hit banks {4n,4n+1}, lanes 16-31 hit {4n+2,4n+3}: 64 banks.
// Each wave computes a full 16x64 row stripe (4 col tiles), so one A-load
// (global b64) feeds 4 WMMAs.  8 waves/block -> 128 rows per block.
__global__ __launch_bounds__(256) void gv_gemm_kernel(
    const float* __restrict__ feat, const float* __restrict__ W,
    const float* __restrict__ bias, float* __restrict__ h, int N) {
    __shared__ float Wq[FDIM * HID];   // 64 KB of the 320 KB/WGP LDS

    // cooperative, coalesced fill of the packed W tile
    for (int s = threadIdx.x; s < FDIM * HID; s += 256) {
        const int kr = s >> 6;    // K row 0..255
        const int n  = s & 63;    // column 0..63
        Wq[((kr >> 2) << 8) + (n << 2) + (kr & 3)] = W[s];
    }
    __syncthreads();

    const int lane = threadIdx.x & 31;
    const int wave = threadIdx.x >> 5;
    const int rowBase = (blockIdx.x * 8 + wave) * 16;
    if (rowBase >= N) return;      // uniform per wave: EXEC all-1s inside

    const int l15   = lane & 15;       // M (for A) / N (for B,C,D)
    const int khalf = (lane >> 4) * 2; // lanes 0-15 -> K 0,1 ; lanes 16-31 -> K 2,3

    const float* arow = feat + (size_t)(rowBase + l15) * FDIM + khalf;
    const float* wbase = Wq + (l15 << 2) + khalf;

    v8f acc[4] = {v8f{}, v8f{}, v8f{}, v8f{}};
#pragma unroll 4
    for (int k = 0; k < FDIM; k += 4) {
        // A tile 16x4 fp32: one b64 per lane, reused by all 4 col tiles
        const v2f a = *(const v2f*)(arow + k);
        // prefetch next K slab of feat (emits global_prefetch_b8)
        __builtin_prefetch(arow + k + 64, 0, 1);
        const float* wq = wbase + (k << 6);  // (k>>2)*256
#pragma unroll
        for (int ct = 0; ct < 4; ++ct) {
            const v2f b = *(const v2f*)(wq + (ct << 6));   // ds_load_b64
            acc[ct] = __builtin_amdgcn_wmma_f32_16x16x4_f32(
                /*neg_a=*/false, a, /*neg_b=*/false, b,
                /*c_mod=*/(short)0, acc[ct], /*reuse_a=*/false, /*reuse_b=*/false);
        }
    }

    const int moff = (lane >> 4) * 8;  // lanes 16-31 hold M = v+8
#pragma unroll
    for (int ct = 0; ct < 4; ++ct) {
        const float bb = bias[ct * 16 + l15];
#pragma unroll
        for (int v = 0; v < 8; ++v) {
            h[(size_t)(rowBase + moff + v) * HID + ct * 16 + l15] = acc[ct][v] + bb;
        }
    }
}

// ---------- kernel 2: agg[row] += v * h[col]   (wave per edge) ----------
__global__ __launch_bounds__(256) void gv_agg_kernel(
    const float* __restrict__ h, const float* __restrict__ vvals,
    const int* __restrict__ row, const int* __restrict__ col,
    float* __restrict__ agg, int E) {
    const int lane = threadIdx.x & 31;
    const int e = blockIdx.x * 8 + (threadIdx.x >> 5);
    if (e >= E) return;
    const int r = row[e], c = col[e];
    const float v = vvals[e];
    const float h0 = h[(size_t)c * HID + lane];
    const float h1 = h[(size_t)c * HID + 32 + lane];
    atomicAdd(&agg[(size_t)r * HID + lane], v * h0);
    atomicAdd(&agg[(size_t)r * HID + 32 + lane], v * h1);
}

// ---------- kernel 3: emb = relu(agg); s_src = emb.w1 ; s_dst = emb.w2 ----------
__global__ __launch_bounds__(256) void gv_node_kernel(
    const float* __restrict__ agg, const float* __restrict__ wmlp,
    float* __restrict__ ssrc, float* __restrict__ sdst, int N) {
    const int lane = threadIdx.x & 31;
    const int n = blockIdx.x * 8 + (threadIdx.x >> 5);
    if (n >= N) return;
    const float e0 = fmaxf(agg[(size_t)n * HID + lane], 0.0f);
    const float e1 = fmaxf(agg[(size_t)n * HID + 32 + lane], 0.0f);
    float a = e0 * wmlp[lane] + e1 * wmlp[32 + lane];
    float b = e0 * wmlp[HID + lane] + e1 * wmlp[HID + 32 + lane];
#pragma unroll
    for (int off = 16; off > 0; off >>= 1) {
        a += __shfl_xor(a, off, 32);
        b += __shfl_xor(b, off, 32);
    }
    if (lane == 0) { ssrc[n] = a; sdst[n] = b; }
}

// ---------- kernel 4: scores[e] + segment-max over row ----------
__global__ void gv_score_kernel(const float* __restrict__ ssrc,
                                const float* __restrict__ sdst,
                                const int* __restrict__ row, const int* __restrict__ col,
                                const float* __restrict__ bmlp,
                                float* __restrict__ scores, int* __restrict__ mkey, int E) {
    const int e = blockIdx.x * blockDim.x + threadIdx.x;
    if (e >= E) return;
    const float sc = ssrc[row[e]] + sdst[col[e]] + bmlp[0];
    scores[e] = sc;
    atomicMax(&mkey[row[e]], f2ord(sc));
}

// ---------- kernel 5: ex = exp(score - m[row]); z[row] += ex ----------
__global__ void gv_exp_kernel(const int* __restrict__ row,
                              const int* __restrict__ mkey,
                              float* __restrict__ scores, float* __restrict__ z, int E) {
    const int e = blockIdx.x * blockDim.x + threadIdx.x;
    if (e >= E) return;
    const int r = row[e];
    const float ex = __expf(scores[e] - ord2f(mkey[r]));
    scores[e] = ex;  // overwrite in place (deterministic)
    atomicAdd(&z[r], ex);
}

// ---------- kernel 6: out = v + lambda * ex / z[row]  (lambda = 1) ----------
__global__ void gv_out_kernel(const float* __restrict__ vvals,
                              const float* __restrict__ ex,
                              const float* __restrict__ z,
                              const int* __restrict__ row,
                              float* __restrict__ out, int E) {
    const int e = blockIdx.x * blockDim.x + threadIdx.x;
    if (e >= E) return;
    out[e] = vvals[e] + ex[e] / z[row[e]];
}

extern "C" void kernel_launch(void* const* d_in, const int* in_sizes, int n_in,
                              void* d_out, int out_size, void* d_ws, size_t ws_size,
                              hipStream_t stream) {
    const float* feat  = (const float*)d_in[0];
    const float* vvals = (const float*)d_in[1];
    const float* Wgcn  = (const float*)d_in[2];
    const float* bgcn  = (const float*)d_in[3];
    const float* wmlp  = (const float*)d_in[4];
    const float* bmlp  = (const float*)d_in[5];
    const int*   vidx  = (const int*)d_in[6];

    const int N = in_sizes[0] / FDIM;   // 100000
    const int E = in_sizes[1];          // 1600000
    const int* row = vidx;
    const int* col = vidx + E;

    // workspace carve-up (floats / ints, all 4-byte)
    float* ws   = (float*)d_ws;
    const size_t N64 = (size_t)N * HID;
    float* h    = ws;                 // N*64
    float* agg  = h + N64;            // N*64
    float* ssrc = agg + N64;          // N
    float* sdst = ssrc + N;           // N
    int*   mkey = (int*)(sdst + N);   // N
    float* z    = (float*)(mkey + N); // N
    float* exv  = z + N;              // E (scores, then exp values)
    float* out  = (float*)d_out;

    // 0) init agg / z / mkey
    {
        int total = N * HID;
        int blocks = (total + 255) / 256;
        gv_init_kernel<<<blocks, 256, 0, stream>>>(agg, z, mkey, N);
    }
    // 1) GEMM: h = feat @ W + b   (WMMA fp32, W staged in LDS)
    {
        int blocks = (N + 127) / 128;   // 128 rows per block (8 waves x 16 rows)
        gv_gemm_kernel<<<blocks, 256, 0, stream>>>(feat, Wgcn, bgcn, h, N);
    }
    // 2) edge aggregation (wave per edge)
    {
        int blocks = (E + 7) / 8;
        gv_agg_kernel<<<blocks, 256, 0, stream>>>(h, vvals, row, col, agg, E);
    }
    // 3) node scores (wave per node)
    {
        int blocks = (N + 7) / 8;
        gv_node_kernel<<<blocks, 256, 0, stream>>>(agg, wmlp, ssrc, sdst, N);
    }
    // 4) edge scores + segment max
    {
        int blocks = (E + 255) / 256;
        gv_score_kernel<<<blocks, 256, 0, stream>>>(ssrc, sdst, row, col, bmlp, exv, mkey, E);
    }
    // 5) exp + segment sum
    {
        int blocks = (E + 255) / 256;
        gv_exp_kernel<<<blocks, 256, 0, stream>>>(row, mkey, exv, z, E);
    }
    // 6) final edge values
    {
        int blocks = (E + 255) / 256;
        gv_out_kernel<<<blocks, 256, 0, stream>>>(vvals, exv, z, row, out, E);
    }
}